// RNN_trial_54649163874757
// MI455X (gfx1250) — compile-verified
//
#include <hip/hip_runtime.h>
#include <math.h>

// Problem constants (from reference): B=64, S=2048, I=H=512, L=3
#define BSZ   64
#define SEQ   2048
#define HID   512
#define NWG   32          // workgroups in the persistent scan; 16 hidden units each
#define UPW   16          // units per workgroup

// LDS weight-slab geometry: 6 tiles x 16 rows x 512 f32, padded by the TDM
// (2 dwords inserted every 256 dwords) -> row stride 516 dwords, slab stride
// 16*516 = 8256 dwords. Stride mod 64 banks = 4 -> conflict-free fragments.
#define SLAB_DW   (16 * 516)
#define LDS_DW    (6 * SLAB_DW)        // 49536 f32 = 198144 B (< 320 KB/WGP)

typedef __attribute__((ext_vector_type(2))) float v2f;
typedef __attribute__((ext_vector_type(8))) float v8f;
typedef unsigned int u32x4 __attribute__((ext_vector_type(4)));
typedef unsigned int u32x8 __attribute__((ext_vector_type(8)));

// ---------------------------------------------------------------------------
// fp32 WMMA: D(16x16) += A(16x4) * B(4x16), full precision matrix-core op.
// A frag: lane holds A[lane%16][k+2*(lane/16) .. +1]  (contiguous f32 pair)
// B frag: lane holds B[k+2*(lane/16)..+1][lane%16]   (pair along K of col n)
// C/D   : vgpr v, lane l -> C[v + 8*(l/16)][l%16]
// ---------------------------------------------------------------------------
__device__ __forceinline__ v8f wmma_f32(v2f a, v2f b, v8f c) {
  return __builtin_amdgcn_wmma_f32_16x16x4_f32(
      /*neg_a=*/false, a, /*neg_b=*/false, b,
      /*c_mod=*/(short)0, c, /*reuse_a=*/false, /*reuse_b=*/false);
}

__device__ __forceinline__ float sigmoidf_(float x) {
  return 1.0f / (1.0f + __expf(-x));
}

// Device-wide barrier: monotonic counter, release/acquire fences around it.
__device__ __forceinline__ void gridbar(unsigned* bar, unsigned target) {
  __threadfence();          // release this thread's global stores
  __syncthreads();
  if (threadIdx.x == 0) {
    __hip_atomic_fetch_add(bar, 1u, __ATOMIC_RELEASE, __HIP_MEMORY_SCOPE_AGENT);
    while (__hip_atomic_load(bar, __ATOMIC_ACQUIRE, __HIP_MEMORY_SCOPE_AGENT) < target)
      __builtin_amdgcn_s_sleep(2);
  }
  __syncthreads();
  __threadfence();          // acquire: see other WGs' stores
}

// ---------------------------------------------------------------------------
// Tensor Data Mover: DMA one 16-row x 512-col f32 tile (a weight slab) from
// global into LDS.  D# built per CDNA5 ISA:
//   group0: count=1 | lds_addr | global_addr | type=2
//   group1: data_size=4B, pad_enable, pad_interval=7 (256 dw), pad_amount=1
//           (2 dw), tensor_dim0=512, tensor_dim1=16, tile_dim0=512,
//           tile_dim1=16, tensor_dim0_stride=512.
// VADDR2/VADDR3 omitted (NULL) -> 2-SGPR-group form. Completion: TENSORcnt.
// ---------------------------------------------------------------------------
__device__ __forceinline__ void tdm_load_slab(unsigned lds_byte_off,
                                              const float* gsrc) {
  const unsigned long long ga = (unsigned long long)(const void*)gsrc;
  u32x4 g0;
  g0[0] = 1u;                                    // count=1 (valid user D#)
  g0[1] = lds_byte_off;                          // LDS destination (bytes)
  g0[2] = (unsigned)(ga & 0xffffffffull);        // global_addr[31:0]
  g0[3] = (unsigned)((ga >> 32) & 0x01ffffffull) // global_addr[56:32]
        | (2u << 30);                            // type=2 ("image")
  u32x8 g1;
  g1[0] = (2u << 16)       // data_size = 4 bytes
        | (1u << 20)       // pad_enable
        | (7u << 22)       // pad_interval: every 256 dwords
        | (1u << 25);      // pad_amount: 2 dwords
  g1[1] = (512u << 16);    // tensor_dim0 = 512 (low 16 in bits 63:48)
  g1[2] = (16u  << 16);    // tensor_dim1 = 16
  g1[3] = (512u << 16);    // tile_dim0 = 512
  g1[4] = 16u;             // tile_dim1 = 16 (tile_dim2 = 0)
  g1[5] = 512u;            // tensor_dim0_stride = 512 (low 32)
  g1[6] = 0u;              // stride high bits / tensor_dim1_stride = 0
  g1[7] = 0u;
  asm volatile("tensor_load_to_lds %0, %1" :: "s"(g0), "s"(g1) : "memory");
}

// ---------------------------------------------------------------------------
// Phase 1: Xpre[s*B+b, :] = x[b,s,:] @ Wi^T + bi + bh      (M=S*B, N=4H, K=I)
// One wave per 16x16 output tile; 8 waves per block.
// ---------------------------------------------------------------------------
__global__ __launch_bounds__(256) void xpre_gemm(
    const float* __restrict__ x, const float* __restrict__ Wi,
    const float* __restrict__ bi, const float* __restrict__ bh,
    float* __restrict__ Xpre)
{
  const int lane = threadIdx.x & 31;
  const int wave = threadIdx.x >> 5;
  const int col  = lane & 15;
  const int hi   = lane >> 4;

  const int mt = blockIdx.x >> 4;                 // 0..8191 M-tiles
  const int nt = ((blockIdx.x & 15) << 3) | wave; // 0..127 N-tiles

  const int m = mt * 16 + col;                    // A row (= s*B + b)
  const int s = m >> 6, b = m & 63;
  const float* arow = x + ((size_t)b * SEQ + s) * HID + 2 * hi;

  const int j = nt * 16 + col;                    // output column (= Wi row)
  const float* brow = Wi + (size_t)j * HID + 2 * hi;

  v8f acc = {0.f, 0.f, 0.f, 0.f, 0.f, 0.f, 0.f, 0.f};
#pragma unroll 8
  for (int k = 0; k < HID; k += 4) {
    v2f a  = *(const v2f*)(arow + k);
    v2f bb = *(const v2f*)(brow + k);
    acc = wmma_f32(a, bb, acc);
  }

  const float bias = bi[j] + bh[j];
#pragma unroll
  for (int v = 0; v < 8; ++v) {
    const int r = mt * 16 + v + 8 * hi;
    Xpre[(size_t)r * (4 * HID) + j] = acc[v] + bias;
  }
}

// ---------------------------------------------------------------------------
// Phase 2: persistent sequential LSTM scan. 32 WGs x 128 threads (4 waves).
// WG w owns hidden units [w*16, w*16+16). Wave wv owns batch rows
// [wv*16, wv*16+16). Gate fragments for a (b,n) element all live in the same
// lane/vgpr slot -> elementwise LSTM math stays in registers; c/m/ws state is
// register-resident for the whole 2048-step loop. Weights are DMA'd once into
// LDS by the TDM; only h/m operands and Xpre touch the memory system per step.
// ---------------------------------------------------------------------------
__global__ __launch_bounds__(128) void lstm_scan(
    const float* __restrict__ x,   const float* __restrict__ h_in,
    const float* __restrict__ c_in,
    const float* __restrict__ Wh,  const float* __restrict__ Wm,
    const float* __restrict__ bm,  const float* __restrict__ Xpre,
    float* __restrict__ out1, float* __restrict__ mbuf,
    unsigned* __restrict__ bar)
{
  __shared__ float wlds[LDS_DW];     // 198144 B of the 320 KB WGP LDS

  const int lane = threadIdx.x & 31;
  const int wv   = threadIdx.x >> 5;
  const int col  = lane & 15;
  const int hi   = lane >> 4;

  const int n0 = blockIdx.x * UPW;   // this WG's hidden-unit range
  const int n  = n0 + col;
  const int b0 = wv * 16;            // this wave's batch-row tile

  // --- one-shot TDM staging of the 6 weight slabs (wave 0 only; EXEC is
  //     ignored by TENSOR ops, the branch keeps other waves from re-issuing)
  if (threadIdx.x < 32) {
    unsigned ldsbase =
        (unsigned)(unsigned long long)(__attribute__((address_space(3))) float*)wlds;
    tdm_load_slab(ldsbase + 0u * SLAB_DW * 4u, Wh + (size_t)(n0)           * HID);
    tdm_load_slab(ldsbase + 1u * SLAB_DW * 4u, Wh + (size_t)(HID + n0)     * HID);
    tdm_load_slab(ldsbase + 2u * SLAB_DW * 4u, Wh + (size_t)(2 * HID + n0) * HID);
    tdm_load_slab(ldsbase + 3u * SLAB_DW * 4u, Wh + (size_t)(3 * HID + n0) * HID);
    tdm_load_slab(ldsbase + 4u * SLAB_DW * 4u, Wm + (size_t)(n0)           * HID);
    tdm_load_slab(ldsbase + 5u * SLAB_DW * 4u, Wm + (size_t)(HID + n0)     * HID);
    __builtin_amdgcn_s_wait_tensorcnt(0);
  }
  __syncthreads();

  // B-fragment LDS bases: row = col (local unit), pair offset 2*hi along K.
  const int lb = col * 516 + 2 * hi;
  const float* L0 = &wlds[0 * SLAB_DW + lb];
  const float* L1 = &wlds[1 * SLAB_DW + lb];
  const float* L2 = &wlds[2 * SLAB_DW + lb];
  const float* L3 = &wlds[3 * SLAB_DW + lb];
  const float* L4 = &wlds[4 * SLAB_DW + lb];
  const float* L5 = &wlds[5 * SLAB_DW + lb];
  const float bm_w = bm[n], bm_g = bm[HID + n];

  // Register-resident state for the 8 (b,n) elements this lane owns.
  float cst[8], mst[8], wst[8];
#pragma unroll
  for (int v = 0; v < 8; ++v) {
    const int b = b0 + v + 8 * hi;
    cst[v] = c_in[(size_t)b * HID + n];   // c[0]
    mst[v] = h_in[(size_t)b * HID + n];   // m0 = h[0]
    wst[v] = -100.0f;
    mbuf[(size_t)b * HID + n] = mst[v];   // slot 0 = initial m
  }
  gridbar(bar, NWG);                      // initial m visible everywhere

  for (int s = 0; s < SEQ; ++s) {
    const float* hA = s ? (out1 + (size_t)(s - 1) * BSZ * HID) : h_in;
    const float* mA = mbuf + (size_t)(s & 1) * BSZ * HID;
    const float* ha_row = hA + (size_t)(b0 + col) * HID + 2 * hi;
    const float* ma_row = mA + (size_t)(b0 + col) * HID + 2 * hi;

    v8f a0 = {0,0,0,0,0,0,0,0}, a1 = a0, a2 = a0, a3 = a0, a4 = a0, a5 = a0;
#pragma unroll 4
    for (int k = 0; k < HID; k += 4) {
      const int kp = k + ((k >> 8) << 1);            // TDM-padded LDS offset
      v2f ah = *(const v2f*)(ha_row + k);
      v2f am = *(const v2f*)(ma_row + k);
      a0 = wmma_f32(ah, *(const v2f*)(L0 + kp), a0);
      a1 = wmma_f32(ah, *(const v2f*)(L1 + kp), a1);
      a2 = wmma_f32(ah, *(const v2f*)(L2 + kp), a2);
      a3 = wmma_f32(ah, *(const v2f*)(L3 + kp), a3);
      a4 = wmma_f32(am, *(const v2f*)(L4 + kp), a4);
      a5 = wmma_f32(am, *(const v2f*)(L5 + kp), a5);
    }

#pragma unroll
    for (int v = 0; v < 8; ++v) {
      const int b = b0 + v + 8 * hi;
      const size_t xr = ((size_t)s * BSZ + b) * (size_t)(4 * HID);
      const float pi = Xpre[xr + n]            + a0[v];
      const float pf = Xpre[xr + HID + n]      + a1[v];
      const float po = Xpre[xr + 2 * HID + n]  + a2[v];
      const float pg = Xpre[xr + 3 * HID + n]  + a3[v];
      const float mw = a4[v] + bm_w;
      const float mg = sigmoidf_(a5[v] + bm_g);
      const float it = sigmoidf_(pi);
      const float ft = sigmoidf_(pf);
      const float ot = sigmoidf_(po + mw);
      const float gt = tanhf(pg);
      const float cn = ft * cst[v] + it * gt + mst[v] * mg;
      cst[v] = cn;
      const float ht = tanhf(cn) * ot;
      const float xt = x[((size_t)b * SEQ + s) * HID + n];
      const bool cond = xt > wst[v];
      const float mn  = cond ? ht : mst[v];
      wst[v] = cond ? xt : wst[v];
      mst[v] = mn;
      out1[(size_t)s * BSZ * HID + (size_t)b * HID + n] = ht;
      mbuf[(size_t)((s + 1) & 1) * BSZ * HID + (size_t)b * HID + n] = mn;
    }
    gridbar(bar, (unsigned)(s + 2) * NWG);
  }
}

// ---------------------------------------------------------------------------
// Phase 3a: G[s,b,g] = out1[s,b,:].Wih3[g,:] + bih3[g];
//           logits[s,b] = out1[s,b,:].m_fin[b,:] / sqrt(H)
// One wave per (s,b); 5 length-512 dots with shuffle reduction.
// ---------------------------------------------------------------------------
__global__ __launch_bounds__(256) void attn_dots(
    const float* __restrict__ out1, const float* __restrict__ mfin,
    const float* __restrict__ Wih3, const float* __restrict__ bih3,
    float* __restrict__ G, float* __restrict__ lg)
{
  const int lane = threadIdx.x & 31;
  const int wave = threadIdx.x >> 5;
  const int g = blockIdx.x * 8 + wave;        // = s*64 + b
  const int b = g & 63;
  const float* orow = out1 + (size_t)g * HID;
  const float* mrow = mfin + (size_t)b * HID;

  float d0 = 0.f, d1 = 0.f, d2 = 0.f, d3 = 0.f, dm = 0.f;
  for (int k = lane; k < HID; k += 32) {
    const float o = orow[k];
    d0 += o * Wih3[k];
    d1 += o * Wih3[HID + k];
    d2 += o * Wih3[2 * HID + k];
    d3 += o * Wih3[3 * HID + k];
    dm += o * mrow[k];
  }
#pragma unroll
  for (int off = 16; off; off >>= 1) {
    d0 += __shfl_xor(d0, off);
    d1 += __shfl_xor(d1, off);
    d2 += __shfl_xor(d2, off);
    d3 += __shfl_xor(d3, off);
    dm += __shfl_xor(dm, off);
  }
  if (lane == 0) {
    G[(size_t)g * 4 + 0] = d0 + bih3[0];
    G[(size_t)g * 4 + 1] = d1 + bih3[1];
    G[(size_t)g * 4 + 2] = d2 + bih3[2];
    G[(size_t)g * 4 + 3] = d3 + bih3[3];
    lg[g] = dm * 0.04419417382415922f;        // 1/sqrt(512)
  }
}

// Phase 3b: hidden-size-1 LSTM scan; one lane per batch element.
__global__ void tiny_scan(
    const float* __restrict__ G, const float* __restrict__ h_in,
    const float* __restrict__ c_in, const float* __restrict__ Whh3,
    const float* __restrict__ bhh3, float* __restrict__ ret)
{
  const int b = threadIdx.x;
  if (b >= BSZ) return;
  float h = h_in[(size_t)2 * BSZ * HID + (size_t)b * HID];  // h[2][b][0]
  float c = c_in[(size_t)2 * BSZ * HID + (size_t)b * HID];
  const float w0 = Whh3[0], w1 = Whh3[1], w2 = Whh3[2], w3 = Whh3[3];
  const float e0 = bhh3[0], e1 = bhh3[1], e2 = bhh3[2], e3 = bhh3[3];
  for (int s = 0; s < SEQ; ++s) {
    const float* gp = G + ((size_t)s * BSZ + b) * 4;
    const float gi = sigmoidf_(gp[0] + h * w0 + e0);
    const float gf = sigmoidf_(gp[1] + h * w1 + e1);
    const float gg = tanhf    (gp[2] + h * w2 + e2);
    const float go = sigmoidf_(gp[3] + h * w3 + e3);
    c = gf * c + gi * gg;
    h = go * tanhf(c);
    ret[(size_t)s * BSZ + b] = h;
  }
}

// Phase 4: per-b softmax over s and weighted sum -> out[b]. 64 blocks.
__global__ __launch_bounds__(256) void softmax_out(
    const float* __restrict__ lg, const float* __restrict__ ret,
    float* __restrict__ out)
{
  __shared__ float red[512];
  const int b = blockIdx.x, t = threadIdx.x;

  float mx = -1e30f;
  for (int s = t; s < SEQ; s += 256) mx = fmaxf(mx, lg[(size_t)s * BSZ + b]);
  red[t] = mx;
  __syncthreads();
  for (int off = 128; off; off >>= 1) {
    if (t < off) red[t] = fmaxf(red[t], red[t + off]);
    __syncthreads();
  }
  mx = red[0];
  __syncthreads();

  float se = 0.f, sr = 0.f;
  for (int s = t; s < SEQ; s += 256) {
    const float e = __expf(lg[(size_t)s * BSZ + b] - mx);
    se += e;
    sr += e * ret[(size_t)s * BSZ + b];
  }
  red[t] = se;
  red[256 + t] = sr;
  __syncthreads();
  for (int off = 128; off; off >>= 1) {
    if (t < off) {
      red[t]       += red[t + off];
      red[256 + t] += red[256 + t + off];
    }
    __syncthreads();
  }
  if (t == 0) out[b] = red[256] / red[0];
}

// ---------------------------------------------------------------------------
extern "C" void kernel_launch(void* const* d_in, const int* in_sizes, int n_in,
                              void* d_out, int out_size, void* d_ws, size_t ws_size,
                              hipStream_t stream) {
  const float* x    = (const float*)d_in[0];
  const float* h    = (const float*)d_in[1];
  const float* c    = (const float*)d_in[2];
  const float* Wi   = (const float*)d_in[3];
  const float* bi   = (const float*)d_in[4];
  const float* Wh   = (const float*)d_in[5];
  const float* bh   = (const float*)d_in[6];
  const float* Wm   = (const float*)d_in[7];
  const float* bm   = (const float*)d_in[8];
  const float* Wih3 = (const float*)d_in[9];
  const float* Whh3 = (const float*)d_in[10];
  const float* bih3 = (const float*)d_in[11];
  const float* bhh3 = (const float*)d_in[12];
  float* out = (float*)d_out;

  // Workspace carve (~1.25 GiB)
  char* w = (char*)d_ws;
  float* Xpre = (float*)w;  w += (size_t)SEQ * BSZ * 4 * HID * sizeof(float); // 1 GiB
  float* out1 = (float*)w;  w += (size_t)SEQ * BSZ * HID * sizeof(float);     // 256 MiB
  float* mbuf = (float*)w;  w += (size_t)2 * BSZ * HID * sizeof(float);
  float* G    = (float*)w;  w += (size_t)SEQ * BSZ * 4 * sizeof(float);
  float* lg   = (float*)w;  w += (size_t)SEQ * BSZ * sizeof(float);
  float* ret  = (float*)w;  w += (size_t)SEQ * BSZ * sizeof(float);
  unsigned* bar = (unsigned*)w;

  hipMemsetAsync(bar, 0, 64, stream);   // reset grid barrier (graph-capture safe)

  // Phase 1: hoisted x@Wi^T GEMM (parallel, WMMA f32)
  xpre_gemm<<<dim3((SEQ * BSZ / 16) * 16), dim3(256), 0, stream>>>(x, Wi, bi, bh, Xpre);
  // Phase 2: persistent recurrent scan (32 WGs, TDM-staged LDS weights,
  //          WMMA f32, grid barrier per step)
  lstm_scan<<<dim3(NWG), dim3(128), 0, stream>>>(x, h, c, Wh, Wm, bm, Xpre, out1, mbuf, bar);
  // Phase 3a: attention logits + layer-3 input gates (final m is mbuf slot 0)
  attn_dots<<<dim3(SEQ * BSZ / 8), dim3(256), 0, stream>>>(out1, mbuf, Wih3, bih3, G, lg);
  // Phase 3b: tiny sequential LSTM (hidden=1)
  tiny_scan<<<dim3(1), dim3(64), 0, stream>>>(G, h, c, Whh3, bhh3, ret);
  // Phase 4: softmax-weighted reduction to out[b]
  softmax_out<<<dim3(BSZ), dim3(256), 0, stream>>>(lg, ret, out);
}